// BalancedErrorRateLoss_43009802502462
// MI455X (gfx1250) — compile-verified
//
#include <hip/hip_runtime.h>

// ---------------------------------------------------------------------------
// BalancedErrorRateLoss for MI455X (gfx1250, wave32).
//
// N = 4,194,304 rows, C = 16 cols, G = 8 groups.
// Memory-bound: ~288 MB streamed -> ~12.4 us at 23.3 TB/s HBM.
// Per-lane register binning for the segment sum; block reduction done as a
// ones-matrix column-sum with V_WMMA_F32_16X16X4_F32 (the problem is a
// one-hot matmul in its original form, so WMMA is the natural reducer).
// ---------------------------------------------------------------------------

typedef __attribute__((ext_vector_type(2))) float v2f;
typedef __attribute__((ext_vector_type(8))) float v8f;
typedef __attribute__((ext_vector_type(4))) int   vi4;

#define NGROUPS 8
#define CCOLS   16
#define BLOCK   256

// --- zero the 16 global accumulators (8 sums, 8 counts) --------------------
__global__ void ws_init_kernel(float* __restrict__ ws) {
  int t = threadIdx.x;
  if (t < 2 * NGROUPS) ws[t] = 0.0f;
}

// --- main pass: gather + |1-x| + per-lane binning + WMMA block reduction ---
__global__ __launch_bounds__(BLOCK)
void ber_partial_kernel(const float* __restrict__ input,
                        const int*   __restrict__ target,
                        const int*   __restrict__ group,
                        float*       __restrict__ ws,
                        int n) {
  // Per-thread row of the block matrix: [8 partial sums | 8 partial counts].
  __shared__ float X[BLOCK * 16];

  float s[NGROUPS];
  float c[NGROUPS];
#pragma unroll
  for (int b = 0; b < NGROUPS; ++b) { s[b] = 0.0f; c[b] = 0.0f; }

  const int  tid     = threadIdx.x;
  const long gtid    = (long)blockIdx.x * BLOCK + tid;
  const long stride4 = (long)gridDim.x * BLOCK * 4;

  // 4 consecutive rows per thread per step: b128 loads for target/group,
  // four b32 gathers from input (each row is 64 B; gathers stream all lines).
  for (long base = gtid * 4; base + 3 < (long)n; base += stride4) {
    vi4 t4 = *(const vi4*)(target + base);
    vi4 g4 = *(const vi4*)(group  + base);
#pragma unroll
    for (int k = 0; k < 4; ++k) {
      float v = input[(base + (long)k) * CCOLS + t4[k]];
      float e = fabsf(1.0f - v);
      int   g = g4[k];
#pragma unroll
      for (int b = 0; b < NGROUPS; ++b) {
        bool m = (g == b);
        s[b] += m ? e    : 0.0f;   // v_cmp + v_cndmask + v_add
        c[b] += m ? 1.0f : 0.0f;   // reuse compare
      }
    }
  }

  // Stage this block's 256x16 partial matrix into LDS.
#pragma unroll
  for (int b = 0; b < NGROUPS; ++b) {
    X[tid * 16 + b]           = s[b];
    X[tid * 16 + NGROUPS + b] = c[b];
  }
  __syncthreads();

  // Wave 0: column-sum the 256x16 matrix with 64 chained WMMAs.
  // A = all-ones (16x4), so D[m][n] = sum_k B[k][n] for the 4 staged rows —
  // invariant to any K-permutation in B's lane layout; only the documented
  // "column n = lane%16" striping and the C/D layout matter.
  if (tid < 32) {
    v2f a;  a[0] = 1.0f; a[1] = 1.0f;
    v8f acc = {};
    const int col  = tid & 15;       // B column n
    const int half = tid >> 4;       // lane half selects K pair {0,1}/{2,3}
#pragma unroll 4
    for (int it = 0; it < 64; ++it) {
      int r = it * 4 + half * 2;     // LDS rows feeding this lane's K slots
      v2f bmat;
      bmat[0] = X[r * 16 + col];
      bmat[1] = X[(r + 1) * 16 + col];
      acc = __builtin_amdgcn_wmma_f32_16x16x4_f32(
          /*neg_a=*/false, a, /*neg_b=*/false, bmat,
          /*c_mod=*/(short)0, acc, /*reuse_a=*/false, /*reuse_b=*/false);
    }
    // C/D layout: VGPR 0, lanes 0-15 = row M=0, cols N=0..15; all rows equal.
    if (tid < 16) atomicAdd(&ws[tid], acc[0]);
  }
}

// --- finalize: guarded means, mean of means, |0.5 - mean| ------------------
__global__ void ber_final_kernel(const float* __restrict__ ws,
                                 float* __restrict__ out) {
  if (threadIdx.x == 0) {
    float msum = 0.0f;
#pragma unroll
    for (int g = 0; g < NGROUPS; ++g) {
      float sg = ws[g];
      float cg = ws[NGROUPS + g];
      float mg = (cg > 0.0f) ? (sg / fmaxf(cg, 1.0f)) : 0.0f;
      msum += mg;
    }
    out[0] = fabsf(0.5f - msum * (1.0f / (float)NGROUPS));
  }
}

extern "C" void kernel_launch(void* const* d_in, const int* in_sizes, int n_in,
                              void* d_out, int out_size, void* d_ws, size_t ws_size,
                              hipStream_t stream) {
  const float* input  = (const float*)d_in[0];   // [N, 16] f32
  const int*   target = (const int*)  d_in[1];   // [N] i32
  const int*   group  = (const int*)  d_in[2];   // [N] i32
  float*       ws     = (float*)d_ws;            // 16 accumulators
  float*       out    = (float*)d_out;           // scalar f32
  const int    n      = in_sizes[1];             // N = 4,194,304

  ws_init_kernel<<<1, 32, 0, stream>>>(ws);

  const int blocks = 2048;                       // 2 grid-stride sweeps over N
  ber_partial_kernel<<<blocks, BLOCK, 0, stream>>>(input, target, group, ws, n);

  ber_final_kernel<<<1, 32, 0, stream>>>(ws, out);
}